// CapsuleLayer_20882130993804
// MI455X (gfx1250) — compile-verified
//
#include <hip/hip_runtime.h>

// CapsuleLayer forward for MI455X (gfx1250, wave32).
//
// Design (bandwidth-roofline driven, see analysis):
//  Kernel 1: u_hat[b,i,o,n] = sum_v inputs[b,i,v] * W[i,o,v,n]
//            via V_WMMA_F32_16X16X4_F32 (f32-exact, K=8 -> 2 WMMA steps).
//            One wave per (i, 16-row b-tile); 10 o-tiles x 2 k-steps = 20 wmma.
//  Kernel 2: fused dynamic routing, one block per batch element.
//            Routing logits + coupling coeffs live in LDS (~92 KB/WG,
//            fits CDNA5's 320 KB WGP LDS). u_hat streamed from L2/HBM.

typedef __attribute__((ext_vector_type(2))) float v2f;
typedef __attribute__((ext_vector_type(8))) float v8f;

#define B_SZ  256
#define I_SZ  1152
#define V_SZ  8
#define O_SZ  10
#define N_SZ  16
#define ON_SZ (O_SZ * N_SZ)   // 160

// ---------------------------------------------------------------------------
// Kernel 1: u_hat GEMM with V_WMMA_F32_16X16X4_F32.
// Per ISA 7.12.2 (wave32):
//   A 16x4 f32 : 2 VGPRs/lane; lanes 0-15 hold (K=0,K=1), lanes 16-31 (K=2,K=3)
//   B 4x16 f32 : 2 VGPRs/lane; same K split, N = lane%16
//   C/D 16x16  : 8 VGPRs; M = vgpr + 8*(lane>=16), N = lane%16
// ---------------------------------------------------------------------------
__global__ __launch_bounds__(32) void uhat_wmma_kernel(
    const float* __restrict__ inp,   // [B, I, V]
    const float* __restrict__ W,     // [I, O, V, N]
    float* __restrict__ uhat)        // [B, I, O*N]
{
    const int lane = threadIdx.x & 31;
    const int half = lane >> 4;      // 0: lanes 0-15, 1: lanes 16-31
    const int l16  = lane & 15;
    const int i    = blockIdx.x;         // input capsule
    const int m0   = blockIdx.y << 4;    // batch-row tile base

    // A fragments for the two K-steps (v = 4*kstep + 2*half + {0,1})
    const float* arow = inp + ((size_t)(m0 + l16) * I_SZ + i) * V_SZ;
    v2f a0, a1;
    a0.x = arow[2 * half + 0];
    a0.y = arow[2 * half + 1];
    a1.x = arow[4 + 2 * half + 0];
    a1.y = arow[4 + 2 * half + 1];

    const float* wbase = W + (size_t)i * O_SZ * V_SZ * N_SZ;

#pragma unroll
    for (int o = 0; o < O_SZ; ++o) {
        const float* wrow = wbase + o * V_SZ * N_SZ;   // [V=8][N=16]
        v2f b0, b1;
        b0.x = wrow[(2 * half + 0) * N_SZ + l16];
        b0.y = wrow[(2 * half + 1) * N_SZ + l16];
        b1.x = wrow[(4 + 2 * half + 0) * N_SZ + l16];
        b1.y = wrow[(4 + 2 * half + 1) * N_SZ + l16];

        v8f c = {};
        c = __builtin_amdgcn_wmma_f32_16x16x4_f32(false, a0, false, b0,
                                                  (short)0, c, false, false);
        c = __builtin_amdgcn_wmma_f32_16x16x4_f32(false, a1, false, b1,
                                                  (short)0, c, false, false);

#pragma unroll
        for (int r = 0; r < 8; ++r) {
            const int m = m0 + r + (half << 3);
            uhat[((size_t)m * I_SZ + i) * ON_SZ + o * N_SZ + l16] = c[r];
        }
    }
}

// ---------------------------------------------------------------------------
// Kernel 2: fused dynamic routing. One block (256 threads) per batch element.
// LDS: blog[I*O] (routing logits), cc[I*O] (coupling coeffs),
//      sv[160] (s then squashed v), scl[16]  -> ~92.8 KB dynamic LDS.
// ---------------------------------------------------------------------------
__global__ __launch_bounds__(256) void routing_kernel(
    const float* __restrict__ uhat,       // [B, I, O*N]
    const int*   __restrict__ n_routing_p,
    float*       __restrict__ out)        // [B, O*N]
{
    extern __shared__ float smem[];
    float* blog = smem;                       // I*O
    float* cc   = blog + I_SZ * O_SZ;         // I*O
    float* sv   = cc + I_SZ * O_SZ;           // 160
    float* scl  = sv + ON_SZ;                 // 16

    const int t  = threadIdx.x;
    const int b  = blockIdx.x;
    const int nr = n_routing_p[0];
    const float* ub = uhat + (size_t)b * I_SZ * ON_SZ;

    for (int idx = t; idx < I_SZ * O_SZ; idx += 256) blog[idx] = 0.0f;
    __syncthreads();

    for (int it = 0; it < nr; ++it) {
        // ---- coupling coefficients: softmax over O per input capsule i ----
        for (int i = t; i < I_SZ; i += 256) {
            float m = blog[i * O_SZ];
#pragma unroll
            for (int o = 1; o < O_SZ; ++o) m = fmaxf(m, blog[i * O_SZ + o]);
            float e[O_SZ];
            float sum = 0.0f;
#pragma unroll
            for (int o = 0; o < O_SZ; ++o) {
                e[o] = __expf(blog[i * O_SZ + o] - m);
                sum += e[o];
            }
            const float inv = 1.0f / sum;
#pragma unroll
            for (int o = 0; o < O_SZ; ++o) cc[i * O_SZ + o] = e[o] * inv;
        }
        __syncthreads();

        // ---- s[o,n] = sum_i cc[i,o] * u_hat[b,i,o,n] ----
        if (t < ON_SZ) {
            const int o = t >> 4;
            const float* up = ub + t;           // stride ON_SZ over i
            float s = 0.0f;
#pragma unroll 4
            for (int i = 0; i < I_SZ; ++i)
                s += cc[i * O_SZ + o] * up[(size_t)i * ON_SZ];
            sv[t] = s;
        }
        __syncthreads();

        // ---- squash: scale = n2 / ((1+n2)*sqrt(n2+eps)) per o ----
        if (t < O_SZ) {
            float n2 = 0.0f;
#pragma unroll
            for (int n = 0; n < N_SZ; ++n) {
                const float x = sv[t * N_SZ + n];
                n2 += x * x;
            }
            scl[t] = n2 / ((1.0f + n2) * sqrtf(n2 + 1e-10f));
        }
        __syncthreads();
        if (t < ON_SZ) sv[t] *= scl[t >> 4];
        __syncthreads();

        // ---- agreement: blog[i,o] += sum_n u_hat[b,i,o,n] * v[o,n] ----
        if (it != nr - 1) {
            const int g = t >> 4;   // 16 groups of 16 lanes (aligned segments)
            const int l = t & 15;
            for (int i = g; i < I_SZ; i += 16) {
                const float* up = ub + (size_t)i * ON_SZ;
#pragma unroll
                for (int o = 0; o < O_SZ; ++o) {
                    float p = up[o * N_SZ + l] * sv[o * N_SZ + l];
                    p += __shfl_xor(p, 1, 16);
                    p += __shfl_xor(p, 2, 16);
                    p += __shfl_xor(p, 4, 16);
                    p += __shfl_xor(p, 8, 16);
                    if (l == 0) blog[i * O_SZ + o] += p;
                }
            }
            __syncthreads();
        }
    }

    if (t < ON_SZ) out[(size_t)b * ON_SZ + t] = sv[t];
}

// ---------------------------------------------------------------------------
extern "C" void kernel_launch(void* const* d_in, const int* in_sizes, int n_in,
                              void* d_out, int out_size, void* d_ws, size_t ws_size,
                              hipStream_t stream) {
    const float* inp = (const float*)d_in[0];   // [256,1152,8] f32
    const float* W   = (const float*)d_in[1];   // [1152,10,8,16] f32
    const int*   nr  = (const int*)d_in[2];     // n_routing (device scalar)
    float* uhat = (float*)d_ws;                 // needs B*I*O*N*4 = 188.7 MB
    float* out  = (float*)d_out;                // [256,10,16] f32

    dim3 g1(I_SZ, B_SZ / 16);
    uhat_wmma_kernel<<<g1, 32, 0, stream>>>(inp, W, uhat);

    const size_t lds_bytes =
        (size_t)(2 * I_SZ * O_SZ + ON_SZ + 16) * sizeof(float);
    routing_kernel<<<B_SZ, 256, lds_bytes, stream>>>(uhat, nr, out);
}